// GLCMModule_28518582845781
// MI455X (gfx1250) — compile-verified
//
#include <hip/hip_runtime.h>
#include <stdint.h>

// ---------------------------------------------------------------------------
// GLCM texture features on MI455X (gfx1250)
//  Kernel 1: gray_pack  : (32,3,8,224,224) f32 -> 256 frames of u8 (in d_ws)
//  Kernel 2: glcm_kernel: 1 workgroup/frame.
//     - frame staged into LDS by the Tensor Data Mover (tensor_load_to_lds)
//     - con/dis/hom streamed (symmetric-weight trick: no histogram needed)
//     - 256x256 u32 histogram in 256KB LDS, built with ds_add_u32
//     - asm = sum(((H+H^T)/(2N))^2) over bins
//     - cross-lane reductions via V_WMMA_F32_16X16X4_F32 (ones-matrix trick)
// ---------------------------------------------------------------------------

typedef float v2f __attribute__((ext_vector_type(2)));
typedef float v8f __attribute__((ext_vector_type(8)));
typedef unsigned int u32x4 __attribute__((ext_vector_type(4)));
typedef unsigned int u32x8 __attribute__((ext_vector_type(8)));

#define HT 224
#define WD 224
#define W4 (WD / 4)              // 56 words per row
#define PIX (HT * WD)            // 50176 pixels / frame
#define PIX4 (PIX / 4)           // 12544 dwords / frame
#define NFRAMES 256              // 32 batch * 8 frames
#define THREADS 1024             // 32 wave32 waves per workgroup

// LDS layout for glcm_kernel (dynamic shared)
#define HIST_WORDS 65536
#define HIST_BYTES (HIST_WORDS * 4)        // 262144
#define FRAME_OFF  HIST_BYTES              // 262144 .. +50176
#define LUT_OFF    (FRAME_OFF + PIX)       // 312320 .. +1024
#define WRED_OFF   (LUT_OFF + 256 * 4)     // 313344 .. +768
#define SMEM_BYTES (WRED_OFF + 32 * 6 * 4) // 314112  (<= 320KB WGP LDS)

// ---------------------------------------------------------------------------
// Full 32-lane all-reduce sum using two V_WMMA_F32_16X16X4_F32.
// WMMA #1: A[16x4] holds val[lane] in K=0 (lanes 0-15) / K=2 (lanes 16-31),
// B = ones -> D[m][n] = val[m] + val[m+16] (replicated over n).
// Per documented C/D layout, lane<16 holds rows 0-7 in d[0..7], lane>=16 rows
// 8-15; summing d[0..7] gives half-sums, WMMA #2 folds the two halves.
// Requires EXEC all ones (call only from fully-converged code).
// ---------------------------------------------------------------------------
__device__ __forceinline__ float wave_sum32(float v) {
  v2f a;    a[0] = v;    a[1] = 0.0f;
  v2f ones; ones[0] = 1.0f; ones[1] = 1.0f;
  v8f c = {};
  v8f d = __builtin_amdgcn_wmma_f32_16x16x4_f32(false, a, false, ones,
                                                (short)0, c, false, false);
  float s = d[0] + d[1] + d[2] + d[3] + d[4] + d[5] + d[6] + d[7];
  v2f a2; a2[0] = s; a2[1] = 0.0f;
  v8f d2 = __builtin_amdgcn_wmma_f32_16x16x4_f32(false, a2, false, ones,
                                                 (short)0, c, false, false);
  return d2[0];
}

// ---------------------------------------------------------------------------
// Kernel 1: grayscale + quantize + byte-pack.  x: [B=32][C=3][F=8][50176]
// One thread per 4 pixels: 3x float4 loads, 1x u32 store. HBM-bound (154MB).
// ---------------------------------------------------------------------------
__global__ void gray_pack(const float* __restrict__ x,
                          uint32_t* __restrict__ gray32, int total4) {
  int t = blockIdx.x * blockDim.x + threadIdx.x;
  if (t >= total4) return;
  int frame = t / PIX4;            // b*8 + f
  int q = t - frame * PIX4;
  int b = frame >> 3;
  int f = frame & 7;
  const float4* base = (const float4*)x;
  float4 r = base[((b * 3 + 0) * 8 + f) * PIX4 + q];
  float4 g = base[((b * 3 + 1) * 8 + f) * PIX4 + q];
  float4 bl = base[((b * 3 + 2) * 8 + f) * PIX4 + q];
  uint32_t out = 0;
#define QUANT(ix, rx, gx, bx)                                         \
  {                                                                   \
    float gray = (rx + gx + bx) / 3.0f;                               \
    int v = (int)floorf(gray * 255.0f);                               \
    v = v < 0 ? 0 : (v > 255 ? 255 : v);                              \
    out |= ((uint32_t)v) << (ix * 8);                                 \
  }
  QUANT(0, r.x, g.x, bl.x)
  QUANT(1, r.y, g.y, bl.y)
  QUANT(2, r.z, g.z, bl.z)
  QUANT(3, r.w, g.w, bl.w)
#undef QUANT
  gray32[t] = out;
}

// ---------------------------------------------------------------------------
// Per-pair processing (a,b byte pair)
// ---------------------------------------------------------------------------
__device__ __forceinline__ void pair_proc(int a, int b, uint32_t* hist,
                                          const float* lut, float& con,
                                          float& dis, float& hom) {
  int d = a - b;
  int ad = d < 0 ? -d : d;
  con += (float)(d * d);
  dis += (float)ad;
  hom += lut[ad];
  atomicAdd(&hist[(a << 8) | b], 1u);   // ds_add_u32
}

// ---------------------------------------------------------------------------
// Per-offset GLCM pass. Word-granular main loop: 4 pairs per (1 a-word +
// 1-2 b-words) LDS loads; ragged edge columns peeled to a small byte pass.
// ---------------------------------------------------------------------------
template <int DR, int DC>
__device__ __forceinline__ void do_offset(const uint32_t* __restrict__ fw,
                                          const unsigned char* __restrict__ frame,
                                          uint32_t* __restrict__ hist,
                                          const float* __restrict__ lut, int tid,
                                          float& conT, float& disT, float& homT,
                                          float& asmT) {
  constexpr int R = HT - DR;                          // rows of pairs
  constexpr int c0 = (DC < 0) ? -DC : 0;
  constexpr int cols = WD - (DC > 0 ? DC : 0) - c0;   // pair columns
  constexpr int N = R * cols;
  constexpr int wlo = (DC < 0) ? 1 : 0;               // full-word range
  constexpr int whi = (DC > 0) ? (W4 - 1) : W4;
  constexpr int NW = whi - wlo;

  // zero histogram (b128 LDS stores)
  {
    uint4 z = make_uint4(0u, 0u, 0u, 0u);
    uint4* h4 = (uint4*)hist;
    for (int i = tid; i < HIST_WORDS / 4; i += THREADS) h4[i] = z;
  }
  __syncthreads();

  float con = 0.f, dis = 0.f, hom = 0.f;

  // main loop: full words only (no masking, no OOB)
  for (int idx = tid; idx < R * NW; idx += THREADS) {
    int r = idx / NW;                   // NW constexpr -> mul/shift
    int w = idx - r * NW + wlo;
    uint32_t aw = fw[r * W4 + w];
    uint32_t bw;
    if (DC == 0) {
      bw = fw[(r + DR) * W4 + w];
    } else if (DC > 0) {
      uint32_t b0 = fw[(r + DR) * W4 + w];
      uint32_t b1 = fw[(r + DR) * W4 + w + 1];
      bw = (b0 >> 8) | (b1 << 24);      // b[k] = frame[r+DR][4w+k+1]
    } else {
      uint32_t b0 = fw[(r + DR) * W4 + w];
      uint32_t bm = fw[(r + DR) * W4 + w - 1];
      bw = (b0 << 8) | (bm >> 24);      // b[k] = frame[r+DR][4w+k-1]
    }
#pragma unroll
    for (int k = 0; k < 4; ++k) {
      pair_proc((aw >> (8 * k)) & 255, (bw >> (8 * k)) & 255, hist, lut, con,
                dis, hom);
    }
  }

  // edge columns (3 per row) for DC != 0
  if (DC != 0) {
    constexpr int ec0 = (DC > 0) ? (whi * 4) : c0;    // 220 or 1
    for (int p = tid; p < R * 3; p += THREADS) {
      int r = p / 3;
      int c = ec0 + (p - r * 3);
      pair_proc(frame[r * WD + c], frame[(r + DR) * WD + c + DC], hist, lut,
                con, dis, hom);
    }
  }
  __syncthreads();

  // asm = sum over all bins of ((H[i,j]+H[j,i]) / (2N))^2
  float asum = 0.f;
  constexpr float inv2N = 1.0f / (2.0f * (float)N);
  for (int bin = tid; bin < HIST_WORDS; bin += THREADS) {
    int i = bin >> 8;
    int j = bin & 255;
    float v = (float)(hist[bin] + hist[(j << 8) | i]) * inv2N;
    asum += v * v;
  }
  __syncthreads();   // histogram free for next offset's zeroing

  constexpr float invN = 1.0f / (float)N;
  conT += con * invN;
  disT += dis * invN;
  homT += hom * invN;
  asmT += asum;
}

// ---------------------------------------------------------------------------
// Kernel 2: one 1024-thread workgroup per frame.
// ---------------------------------------------------------------------------
__global__ void __launch_bounds__(THREADS, 1)
glcm_kernel(const uint32_t* __restrict__ gray32, float* __restrict__ out) {
  extern __shared__ unsigned char smem[];
  uint32_t* hist = (uint32_t*)smem;
  unsigned char* frame = smem + FRAME_OFF;
  const uint32_t* fw = (const uint32_t*)frame;
  float* lut = (float*)(smem + LUT_OFF);
  float* wred = (float*)(smem + WRED_OFF);

  const int tid = threadIdx.x;
  const int frame_idx = blockIdx.x;

  // -------------------------------------------------------------------
  // Stage the 50176-byte frame into LDS with the Tensor Data Mover.
  // D# (CDNA5 ISA): 2-group descriptor, 2D tensor,
  // data_size=1B, tensor_dim0 = tile_dim0 = 50176, tile_dim1 = 1.
  // Issued by wave 0 only behind a *scalar* branch (TDM ignores EXEC, so an
  // exec-masked guard would duplicate the DMA per wave).
  // -------------------------------------------------------------------
  if (__builtin_amdgcn_readfirstlane(tid) < 32) {
    uint64_t ga = (uint64_t)(uintptr_t)(gray32 + (size_t)frame_idx * PIX4);
    u32x4 g0;
    g0[0] = 1u;                                   // count=1, user desc
    g0[1] = (unsigned)FRAME_OFF;                  // lds_addr
    g0[2] = (uint32_t)ga;                         // global_addr[31:0]
    g0[3] = (uint32_t)((ga >> 32) & 0x1FFFFFFu)   // global_addr[56:32]
            | (2u << 30);                         // type = 2 ("image")
    u32x8 g1;
    g1[0] = 0u;                                   // no multicast, data_size=1B
    g1[1] = ((uint32_t)PIX & 0xFFFFu) << 16;      // tensor_dim0[15:0]
    g1[2] = ((uint32_t)PIX >> 16)                 // tensor_dim0[31:16]
            | (1u << 16);                         // tensor_dim1[15:0] = 1
    g1[3] = ((uint32_t)PIX & 0xFFFFu) << 16;      // tile_dim0 = 50176
    g1[4] = 1u;                                   // tile_dim1 = 1, tile_dim2=0
    g1[5] = (uint32_t)PIX;                        // tensor_dim0_stride[31:0]
    g1[6] = ((uint32_t)PIX & 0xFFFFu) << 16;      // dim1_stride[15:0]
    g1[7] = 0u;                                   // dim1_stride[47:16]
    asm volatile("tensor_load_to_lds %0, %1" :: "s"(g0), "s"(g1) : "memory");
    __builtin_amdgcn_s_wait_tensorcnt(0);
  }
  // homogeneity LUT: 1/(1+d^2)
  if (tid < 256) lut[tid] = 1.0f / (1.0f + (float)(tid * tid));
  __syncthreads();   // frame (TDM complete in wave 0) + LUT visible to all

  // std sums from packed bytes
  float sg = 0.f, sg2 = 0.f;
  for (int i = tid; i < PIX4; i += THREADS) {
    uint32_t w = fw[i];
    int b0 = w & 255, b1 = (w >> 8) & 255, b2 = (w >> 16) & 255, b3 = w >> 24;
    sg += (float)(b0 + b1 + b2 + b3);
    sg2 += (float)(b0 * b0 + b1 * b1 + b2 * b2 + b3 * b3);
  }

  float conT = 0.f, disT = 0.f, homT = 0.f, asmT = 0.f;
  do_offset<0, 1>(fw, frame, hist, lut, tid, conT, disT, homT, asmT);
  do_offset<1, 1>(fw, frame, hist, lut, tid, conT, disT, homT, asmT);
  do_offset<1, 0>(fw, frame, hist, lut, tid, conT, disT, homT, asmT);
  do_offset<1, -1>(fw, frame, hist, lut, tid, conT, disT, homT, asmT);

  // two-level reduction: WMMA 32-lane all-reduce, then wave0 over 32 partials
  const int lane = tid & 31;
  const int wave = tid >> 5;
  float vals[6] = {sg, sg2, conT, disT, homT, asmT};
#pragma unroll
  for (int k = 0; k < 6; ++k) {
    float t = wave_sum32(vals[k]);
    if (lane == 0) wred[wave * 6 + k] = t;
  }
  __syncthreads();

  if (wave == 0) {
    float tot[6];
#pragma unroll
    for (int k = 0; k < 6; ++k) tot[k] = wave_sum32(wred[lane * 6 + k]);
    if (lane == 0) {
      const float invP = 1.0f / (float)PIX;
      float mean = tot[0] * invP;
      float var = tot[1] * invP - mean * mean;
      float stdv = sqrtf(fmaxf(var, 0.0f));
      float con = tot[2] * 0.25f;
      float dis = tot[3] * 0.25f;
      float hom = tot[4] * 0.25f;
      float as = tot[5] * 0.25f;
      float en = sqrtf(as);
      float* o = out + (size_t)frame_idx * 6;
      o[0] = stdv; o[1] = con; o[2] = dis; o[3] = hom; o[4] = as; o[5] = en;
    }
  }
}

// ---------------------------------------------------------------------------
extern "C" void kernel_launch(void* const* d_in, const int* in_sizes, int n_in,
                              void* d_out, int out_size, void* d_ws,
                              size_t ws_size, hipStream_t stream) {
  (void)in_sizes; (void)n_in; (void)out_size; (void)ws_size;
  const float* x = (const float*)d_in[0];
  float* out = (float*)d_out;
  uint32_t* gray = (uint32_t*)d_ws;   // 256 frames * 12544 dwords = 12.8 MB

  // opt into >64KB dynamic LDS (314112 B; CDNA5 WGP has 320KB)
  hipFuncSetAttribute((const void*)glcm_kernel,
                      hipFuncAttributeMaxDynamicSharedMemorySize, SMEM_BYTES);

  const int total4 = NFRAMES * PIX4;  // 3,211,264
  gray_pack<<<(total4 + 255) / 256, 256, 0, stream>>>(x, gray, total4);
  glcm_kernel<<<NFRAMES, THREADS, SMEM_BYTES, stream>>>(gray, out);
}